// Patch_QML_Engine_47167330844881
// MI455X (gfx1250) — compile-verified
//
#include <hip/hip_runtime.h>
#include <hip/hip_bf16.h>
#include <math.h>

// ---------------------------------------------------------------------------
// Patch_QML_Engine for MI455X (gfx1250)
//
// Key algebraic collapse: the StronglyEntanglingLayers circuit has patch-
// independent parameters, so it is a constant 16x16 complex unitary U.
//   enc_q(psi) = sum_k s[k][q] * ((ReU . psi)_k^2 + (ImU . psi)_k^2)
// => 1) build W = [ReU; ImU] (32x16 f32) once
//    2) Y = Psi @ W^T via V_WMMA_F32_16X16X4_F32 (full fp32 precision)
//    3) out = enc @ head_w^T + b via the same WMMA shape, K = 500 = 125*4
// Total ~1 GFLOP, all data L2-resident: latency-bound, so we keep native
// f32 WMMA (no precision loss vs the f32 reference).
// ---------------------------------------------------------------------------

typedef __attribute__((ext_vector_type(2))) float v2f;
typedef __attribute__((ext_vector_type(8))) float v8f;

#define SEQ_LEN     512
#define PRED_LEN    96
#define NUM_PATCHES 125
#define NPQ         500      // NUM_PATCHES * 4
#define BM          4096     // 64 batch * 64 channels

// ---------------------------------------------------------------------------
// Kernel 1: build W = [Re U ; Im U] (32x16, row-major) from circuit weights.
// Thread j (j<16) evolves basis column e_j through 2 StronglyEntangling
// layers: Rot(phi,theta,omega) per qubit (wire0 = MSB of 4-bit index) then
// CNOT ring with range r = (layer % 3) + 1. Fully unrolled -> registers.
// ---------------------------------------------------------------------------
__global__ void build_unitary_kernel(const float* __restrict__ weights,
                                     float* __restrict__ W)
{
    const int j = threadIdx.x;
    if (j >= 16) return;

    float wr[16], wi[16];
#pragma unroll
    for (int k = 0; k < 16; ++k) { wr[k] = (k == j) ? 1.f : 0.f; wi[k] = 0.f; }

#pragma unroll
    for (int l = 0; l < 2; ++l) {
#pragma unroll
        for (int q = 0; q < 4; ++q) {
            const float phi = weights[(l * 4 + q) * 3 + 0];
            const float th  = weights[(l * 4 + q) * 3 + 1];
            const float om  = weights[(l * 4 + q) * 3 + 2];
            const float c  = cosf(0.5f * th),  s  = sinf(0.5f * th);
            const float hf = 0.5f * (phi + om), df = 0.5f * (phi - om);
            const float ch = cosf(hf), sh = sinf(hf);
            const float cd = cosf(df), sd = sinf(df);
            // U = [[e^{-i hf} c, -e^{+i df} s], [e^{-i df} s, e^{+i hf} c]]
            const float u00r =  ch * c, u00i = -sh * c;
            const float u01r = -cd * s, u01i = -sd * s;
            const float u10r =  cd * s, u10i = -sd * s;
            const float u11r =  ch * c, u11i =  sh * c;
            const int st = 1 << (3 - q);   // wire0 = MSB
#pragma unroll
            for (int k = 0; k < 16; ++k) {
                if ((k & st) == 0) {
                    const int k1 = k | st;
                    const float a0r = wr[k],  a0i = wi[k];
                    const float a1r = wr[k1], a1i = wi[k1];
                    wr[k]  = u00r*a0r - u00i*a0i + u01r*a1r - u01i*a1i;
                    wi[k]  = u00r*a0i + u00i*a0r + u01r*a1i + u01i*a1r;
                    wr[k1] = u10r*a0r - u10i*a0i + u11r*a1r - u11i*a1i;
                    wi[k1] = u10r*a0i + u10i*a0r + u11r*a1i + u11i*a1r;
                }
            }
        }
        const int r = (l % 3) + 1;
#pragma unroll
        for (int i = 0; i < 4; ++i) {
            const int t = (i + r) & 3;
            const int cm = 1 << (3 - i), tm = 1 << (3 - t);
#pragma unroll
            for (int k = 0; k < 16; ++k) {
                if ((k & cm) && !(k & tm)) {
                    const int k1 = k | tm;
                    float tr = wr[k]; wr[k] = wr[k1]; wr[k1] = tr;
                    float ti = wi[k]; wi[k] = wi[k1]; wi[k1] = ti;
                }
            }
        }
    }
#pragma unroll
    for (int k = 0; k < 16; ++k) {
        W[k * 16 + j]        = wr[k];   // Re U[k][j]
        W[(k + 16) * 16 + j] = wi[k];   // Im U[k][j]
    }
}

// ---------------------------------------------------------------------------
// Kernel 2: patch encoder. One wave handles one (series, 16-patch tile):
//   load 80 series elements -> LDS, per-patch 1/||.||, build f32 A fragment,
//   Yr/Yi via 4+4 V_WMMA_F32_16X16X4_F32, probs = Yr^2+Yi^2, LDS signed
//   reduction to enc[bm][n*4+q].
// A layout (16x4 f32): lane L holds A[L%16][2*(L/16)+v] in VGPR v.
// B layout (4x16 f32): lane L holds B[2*(L/16)+v][L%16] in VGPR v.
// C/D layout: lane L, VGPR v -> M = v + 8*(L/16), N = L%16.
// ---------------------------------------------------------------------------
#define ENC_WAVES 4
__global__ __launch_bounds__(32 * ENC_WAVES)
void encode_patches_kernel(const float* __restrict__ x,
                           const float* __restrict__ W,
                           float* __restrict__ enc)
{
    __shared__ float lds[ENC_WAVES][352];  // 80 buf + 16 invn + 256 ptile
    const int w = threadIdx.x >> 5;
    const int L = threadIdx.x & 31;
    const int unit = blockIdx.x * ENC_WAVES + w;    // 0 .. 32767
    const int bm   = unit >> 3;
    const int n0   = (unit & 7) << 4;               // patch tile base
    const int b    = bm >> 6, m = bm & 63;

    const float* series = x + (size_t)b * (SEQ_LEN * 64) + m;  // elem l at l*64
    float* buf   = lds[w];        // [80] series window (+1e-6 applied)
    float* invn  = buf + 80;      // [16] per-patch 1/norm
    float* ptile = buf + 96;      // [256] probs tile (16m x 16k)

    for (int i = L; i < 80; i += 32) {
        int l = 4 * n0 + i; if (l > SEQ_LEN - 1) l = SEQ_LEN - 1;  // pad clamp
        buf[i] = series[(size_t)l * 64] + 1e-6f;
    }
    __syncthreads();

    if (L < 16) {
        float s = 0.f;
#pragma unroll
        for (int jj = 0; jj < 16; ++jj) { const float v = buf[4 * L + jj]; s += v * v; }
        invn[L] = 1.0f / sqrtf(s);
    }
    __syncthreads();

    const int   lrow  = L & 15;
    const int   kbase = (L >> 4) << 1;
    const float scl   = invn[lrow];

    v8f cr = {0.f,0.f,0.f,0.f,0.f,0.f,0.f,0.f};
    v8f ci = {0.f,0.f,0.f,0.f,0.f,0.f,0.f,0.f};
#pragma unroll
    for (int kk = 0; kk < 4; ++kk) {
        const int j0 = 4 * kk + kbase;
        v2f a  = { buf[4 * lrow + j0] * scl, buf[4 * lrow + j0 + 1] * scl };
        v2f br = { W[lrow * 16 + j0],        W[lrow * 16 + j0 + 1] };        // Re U
        v2f bi = { W[(lrow + 16) * 16 + j0], W[(lrow + 16) * 16 + j0 + 1] }; // Im U
        cr = __builtin_amdgcn_wmma_f32_16x16x4_f32(false, a, false, br, (short)0, cr, false, false);
        ci = __builtin_amdgcn_wmma_f32_16x16x4_f32(false, a, false, bi, (short)0, ci, false, false);
    }

    const int mhi = (L >> 4) << 3;
#pragma unroll
    for (int v = 0; v < 8; ++v) {
        const float p = cr[v] * cr[v] + ci[v] * ci[v];   // |amp|^2
        ptile[(v + mhi) * 16 + lrow] = p;
    }
    __syncthreads();

    // <Z_q> = sum_k p_k * (1 - 2*bit_{3-q}(k)); 64 outputs, 2 per lane.
#pragma unroll
    for (int o = L; o < 64; o += 32) {
        const int mo = o >> 2, q = o & 3;
        const int n = n0 + mo;
        if (n < NUM_PATCHES) {
            float s = 0.f;
#pragma unroll
            for (int k = 0; k < 16; ++k) {
                const float sg = ((k >> (3 - q)) & 1) ? -1.f : 1.f;
                s += ptile[mo * 16 + k] * sg;
            }
            enc[(size_t)bm * NPQ + n * 4 + q] = s;
        }
    }
}

// ---------------------------------------------------------------------------
// Kernel 3: head GEMM. out[4096 x 96] = enc[4096 x 500] @ head_w^T + b,
// then scatter to out[b][p][m]. One wave per 16x16 tile, 125 f32 WMMAs.
// ---------------------------------------------------------------------------
#define HEAD_WAVES 4
__global__ __launch_bounds__(32 * HEAD_WAVES)
void head_gemm_kernel(const float* __restrict__ enc,
                      const float* __restrict__ head_w,
                      const float* __restrict__ head_b,
                      float* __restrict__ out)
{
    const int w = threadIdx.x >> 5;
    const int L = threadIdx.x & 31;
    const int u = blockIdx.x * HEAD_WAVES + w;      // 0 .. 1535
    const int mt = u / 6, nt = u - mt * 6;
    const int row0 = mt << 4, col0 = nt << 4;

    const int lrow  = L & 15;
    const int kbase = (L >> 4) << 1;
    const int pcol  = col0 + lrow;

    const float* arow = enc    + (size_t)(row0 + lrow) * NPQ;  // A[M=lrow][.]
    const float* brow = head_w + (size_t)pcol * NPQ;           // B[j][N=lrow]

    v8f acc = {0.f,0.f,0.f,0.f,0.f,0.f,0.f,0.f};
    for (int kk = 0; kk < 125; ++kk) {
        const int j0 = 4 * kk + kbase;
        v2f a = *(const v2f*)(arow + j0);   // 8B-aligned (NPQ*4 % 8 == 0)
        v2f bb = *(const v2f*)(brow + j0);
        acc = __builtin_amdgcn_wmma_f32_16x16x4_f32(false, a, false, bb, (short)0, acc, false, false);
    }

    const float bias = head_b[pcol];
    const int mhi = (L >> 4) << 3;
#pragma unroll
    for (int v = 0; v < 8; ++v) {
        const int row  = row0 + v + mhi;        // bm
        const int bidx = row >> 6, midx = row & 63;
        out[((size_t)bidx * PRED_LEN + pcol) * 64 + midx] = acc[v] + bias;
    }
}

// ---------------------------------------------------------------------------
extern "C" void kernel_launch(void* const* d_in, const int* in_sizes, int n_in,
                              void* d_out, int out_size, void* d_ws, size_t ws_size,
                              hipStream_t stream)
{
    const float* x       = (const float*)d_in[0];  // [64, 512, 64]
    const float* weights = (const float*)d_in[1];  // [2, 4, 3]
    const float* head_w  = (const float*)d_in[2];  // [96, 500]
    const float* head_b  = (const float*)d_in[3];  // [96]
    float*       out     = (float*)d_out;          // [64, 96, 64]

    float* W   = (float*)d_ws;                     // 32*16 floats
    float* enc = W + 512;                          // 4096*500 floats (~8 MB)

    build_unitary_kernel<<<1, 32, 0, stream>>>(weights, W);

    // 4096 series * 8 patch-tiles = 32768 waves, 4 waves/block
    encode_patches_kernel<<<(BM * 8) / ENC_WAVES, 32 * ENC_WAVES, 0, stream>>>(x, W, enc);

    // 256 M-tiles * 6 N-tiles = 1536 waves, 4 waves/block
    head_gemm_kernel<<<(256 * 6) / HEAD_WAVES, 32 * HEAD_WAVES, 0, stream>>>(enc, head_w, head_b, out);
}